// HungarianMatcher_11175504904384
// MI455X (gfx1250) — compile-verified
//
#include <hip/hip_runtime.h>
#include <cstddef>

// ---------------------------------------------------------------------------
// DETR Hungarian matcher cost matrix on MI455X (gfx1250, wave32).
//
//   C[n,m] = L1(pred_box[n], tgt_box[m])
//          + K[n, tgt_ids[m]]                (focal class cost, via WMMA)
//          - GIoU(pred_box[n], tgt_box[m])
//
// Class-cost gather expressed as C_class = K[N,96] x OneHot[96,M] using
// V_WMMA_F32_16X16X4_F32 (exact: one-hot B => no accumulation error).
//
// Roofline: 92 MB output write -> ~4 us floor @ 23.3 TB/s HBM. Epilogue is
// kept at ~32 VALU ops/element (v_rcp instead of IEEE div; pred-side xyxy
// hoisted out of the column sweep) so VALU stays at the memory floor. Each
// wave holds its 16x96 A strip in registers across 5 column tiles (5x less
// L2 read traffic on the K matrix).
// ---------------------------------------------------------------------------

typedef __attribute__((ext_vector_type(2))) float v2f;
typedef __attribute__((ext_vector_type(8))) float v8f;

#define BS_   16
#define Q_    900
#define NROW  (BS_ * Q_)      // 14400 rows (bs*num_queries)
#define NCLS  91
#define KPAD  96              // class dim padded to multiple of 4 (WMMA K step)
#define MTGT  1600            // number of targets
#define CTILES_PER_WAVE 5     // column tiles per wave (A-strip reuse factor)
#define CGROUPS ((MTGT / 16) / CTILES_PER_WAVE)   // 20 column groups
#define FEPS  1e-8f

// ---------------------------------------------------------------------------
// Kernel 1: K[n,c] = focal(pos) - focal(neg) for c < 91, else 0.
// 5.5 MB -> stays resident in the 192 MB L2.
// ---------------------------------------------------------------------------
__global__ __launch_bounds__(256)
void focal_precompute(const float* __restrict__ logits, float* __restrict__ Kmat)
{
    int idx = blockIdx.x * 256 + threadIdx.x;
    if (idx >= NROW * KPAD) return;
    int c = idx % KPAD;
    int n = idx / KPAD;
    float k = 0.0f;
    if (c < NCLS) {
        float x = logits[n * NCLS + c];
        float p = 1.0f / (1.0f + __expf(-x));          // sigmoid
        // alpha=0.25, gamma=2
        float neg = 0.75f * p * p * (-__logf(1.0f - p + FEPS));
        float pos = 0.25f * (1.0f - p) * (1.0f - p) * (-__logf(p + FEPS));
        k = pos - neg;
    }
    Kmat[idx] = k;
}

// ---------------------------------------------------------------------------
// Kernel 2: each wave32 owns a 16-row x 80-col slab of C (5 column tiles).
//   - A strip (16 rows x 96 classes, f32) loaded ONCE into 24 x v2f regs
//   - pred-side xyxy/area hoisted once per row into registers
//   - per column tile: one-hot B from integer compares, 24x WMMA, fused
//     L1 + GIoU epilogue (v_rcp-based divides), store
// A layout (ISA 7.12.2): lanes 0-15 hold K=k,k+1 ; lanes 16-31 K=k+2,k+3,
//   row = lane&15.
// C/D layout: VGPR r, lane L -> row r + 8*(L>=16), col L&15.
// ---------------------------------------------------------------------------
__global__ __launch_bounds__(256)
void cost_matrix(const float* __restrict__ Kmat,
                 const float* __restrict__ pred_boxes,
                 const int*   __restrict__ tgt_ids,
                 const float* __restrict__ tgt_bbox,
                 float*       __restrict__ out)
{
    const int lane = threadIdx.x & 31;
    const int wave = threadIdx.x >> 5;
    const int task = blockIdx.x * 8 + wave;      // 18000 tasks, no tail
    const int row_base  = (task / CGROUPS) * 16;
    const int col_base0 = (task % CGROUPS) * (16 * CTILES_PER_WAVE);

    const int l15   = lane & 15;
    const int khalf = (lane >> 4) << 1;          // 0 for lanes 0-15, 2 for 16-31

    // -------- load full A strip into registers (reused for 5 column tiles) -
    const float* arow = Kmat + (size_t)(row_base + l15) * KPAD + khalf;
    __builtin_prefetch(arow, 0, 0);              // global_prefetch_b8
    v2f A[KPAD / 4];
#pragma unroll
    for (int i = 0; i < KPAD / 4; ++i)
        A[i] = *(const v2f*)(arow + 4 * i);      // global_load_b64

    // -------- load this lane's 8 pred boxes + hoist derived values ---------
    const int row0 = row_base + ((lane >> 4) << 3);
    float4 pb[8];
    float px0[8], py0[8], px1[8], py1[8], pa[8];
#pragma unroll
    for (int r = 0; r < 8; ++r) {
        pb[r] = reinterpret_cast<const float4*>(pred_boxes)[row0 + r];
        px0[r] = pb[r].x - 0.5f * pb[r].z;
        py0[r] = pb[r].y - 0.5f * pb[r].w;
        px1[r] = pb[r].x + 0.5f * pb[r].z;
        py1[r] = pb[r].y + 0.5f * pb[r].w;
        pa[r]  = pb[r].z * pb[r].w;
    }

    // -------- sweep 5 column tiles ----------------------------------------
    for (int t = 0; t < CTILES_PER_WAVE; ++t) {
        const int col = col_base0 + t * 16 + l15;
        const int rel = tgt_ids[col] - khalf;    // one-hot: b.x=(rel==kc), b.y=(rel==kc+1)

        v8f acc = {};
#pragma unroll
        for (int kc = 0; kc < KPAD; kc += 4) {
            v2f b;
            b.x = (rel == kc)     ? 1.0f : 0.0f;
            b.y = (rel == kc + 1) ? 1.0f : 0.0f;
            // (neg_a, A, neg_b, B, c_mod, C, reuse_a, reuse_b)
            acc = __builtin_amdgcn_wmma_f32_16x16x4_f32(
                false, A[kc >> 2], false, b, (short)0, acc, false, false);
        }

        // ---- target-side values (once per tile) ---------------------------
        const float4 tb = reinterpret_cast<const float4*>(tgt_bbox)[col]; // cx,cy,w,h
        const float tx0 = tb.x - 0.5f * tb.z, ty0 = tb.y - 0.5f * tb.w;
        const float tx1 = tb.x + 0.5f * tb.z, ty1 = tb.y + 0.5f * tb.w;
        const float area_t = tb.z * tb.w;

        // ---- fused L1 + GIoU epilogue + store -----------------------------
#pragma unroll
        for (int r = 0; r < 8; ++r) {
            // L1 over cxcywh (abs folds into source modifiers)
            float l1 = __builtin_fabsf(pb[r].x - tb.x) + __builtin_fabsf(pb[r].y - tb.y)
                     + __builtin_fabsf(pb[r].z - tb.z) + __builtin_fabsf(pb[r].w - tb.w);

            // GIoU on xyxy; divides via v_rcp_f32 (TRANS, co-executes w/ WMMA)
            const float iw = fmaxf(fminf(px1[r], tx1) - fmaxf(px0[r], tx0), 0.0f);
            const float ih = fmaxf(fminf(py1[r], ty1) - fmaxf(py0[r], ty0), 0.0f);
            const float inter = iw * ih;
            const float uni   = pa[r] + area_t - inter;
            const float iou   = inter * __builtin_amdgcn_rcpf(uni);

            const float ew = fmaxf(fmaxf(px1[r], tx1) - fminf(px0[r], tx0), 0.0f);
            const float eh = fmaxf(fmaxf(py1[r], ty1) - fminf(py0[r], ty0), 0.0f);
            const float ae = ew * eh;
            const float giou = iou - (ae - uni) * __builtin_amdgcn_rcpf(ae);

            out[(size_t)(row0 + r) * MTGT + col] = l1 + acc[r] - giou;
        }
    }
}

// ---------------------------------------------------------------------------
// Host-side launcher (graph-capture safe: only kernel launches on `stream`).
// ---------------------------------------------------------------------------
extern "C" void kernel_launch(void* const* d_in, const int* in_sizes, int n_in,
                              void* d_out, int out_size, void* d_ws, size_t ws_size,
                              hipStream_t stream)
{
    const float* pred_logits = (const float*)d_in[0];  // [16,900,91] f32
    const float* pred_boxes  = (const float*)d_in[1];  // [16,900,4]  f32
    const int*   tgt_ids     = (const int*)  d_in[2];  // [1600] int32
    const float* tgt_bbox    = (const float*)d_in[3];  // [1600,4] f32
    float*       out         = (float*)d_out;          // [16,900,1600] f32
    float*       Kmat        = (float*)d_ws;           // [14400,96] f32 = 5.5 MB

    // Kernel 1: focal class-cost matrix (padded to 96 columns)
    {
        int total  = NROW * KPAD;
        int blocks = (total + 255) / 256;
        hipLaunchKernelGGL(focal_precompute, dim3(blocks), dim3(256), 0, stream,
                           pred_logits, Kmat);
    }
    // Kernel 2: fused WMMA one-hot GEMM + L1 + GIoU
    {
        int tasks  = (NROW / 16) * CGROUPS;      // 18000 wave tasks
        int blocks = tasks / 8;                  // 2250 blocks, 8 waves each
        hipLaunchKernelGGL(cost_matrix, dim3(blocks), dim3(256), 0, stream,
                           Kmat, pred_boxes, tgt_ids, tgt_bbox, out);
    }
}